// SAGE_45784351375947
// MI455X (gfx1250) — compile-verified
//
#include <hip/hip_runtime.h>
#include <math.h>

typedef float v2f __attribute__((ext_vector_type(2)));
typedef float v8f __attribute__((ext_vector_type(8)));

// ---------------------------------------------------------------------------
// Zero-fill (grid-stride)
// ---------------------------------------------------------------------------
__global__ void fill_zero_f32(float* __restrict__ p, long n) {
    long i = (long)blockIdx.x * blockDim.x + threadIdx.x;
    long stride = (long)gridDim.x * blockDim.x;
    for (; i < n; i += stride) p[i] = 0.0f;
}

// ---------------------------------------------------------------------------
// Edge scatter: one wave32 per edge; fp32 hardware atomics into agg + cnt
// ---------------------------------------------------------------------------
__global__ __launch_bounds__(256) void sage_scatter(
    const float* __restrict__ Xsrc, const int* __restrict__ src,
    const int* __restrict__ dst, float* __restrict__ agg,
    float* __restrict__ cnt, int nE, int fin)
{
    const int lane = threadIdx.x & 31;
    const int e = blockIdx.x * (blockDim.x >> 5) + (threadIdx.x >> 5);
    if (e >= nE) return;
    const int s = src[e];
    const int d = dst[e];
    const float* xs = Xsrc + (size_t)s * fin;
    float* ag = agg + (size_t)d * fin;
    for (int k = lane; k < fin; k += 32) {
        __hip_atomic_fetch_add(&ag[k], xs[k], __ATOMIC_RELAXED,
                               __HIP_MEMORY_SCOPE_AGENT);
    }
    if (lane == 0) {
        __hip_atomic_fetch_add(&cnt[d], 1.0f, __ATOMIC_RELAXED,
                               __HIP_MEMORY_SCOPE_AGENT);
    }
}

// ---------------------------------------------------------------------------
// agg[r][:] /= max(cnt[r], 1)
// ---------------------------------------------------------------------------
__global__ void sage_normalize(float* __restrict__ agg,
                               const float* __restrict__ cnt,
                               long total, int fin)
{
    long i = (long)blockIdx.x * blockDim.x + threadIdx.x;
    long stride = (long)gridDim.x * blockDim.x;
    for (; i < total; i += stride) {
        long r = i / fin;
        float s = 1.0f / fmaxf(cnt[r], 1.0f);
        agg[i] *= s;
    }
}

// ---------------------------------------------------------------------------
// Fused SAGE GEMM: Out = relu?( Agg@Wl + Xdst@Wr + bias )
//
// One wave per 16x(16*NT) output strip: NT accumulators reuse each A fragment
// for NT WMMAs (doubled arithmetic intensity on the A stream for NT=2).
// V_WMMA_F32_16X16X4_F32, K stepped by 4. The 8 waves of a block share one
// group of NT column tiles (tilesM multiple of 8 => group id block-uniform);
// Wl/Wr column tiles are staged once into LDS (coalesced) and read
// conflict-free. Staging columns are CLAMPED (not guarded) so the compute
// path is branch-free with EXEC all-ones (WMMA requirement); stores mask.
//
// fp32 fragment layouts (ISA 7.12.2):
//   A 16x4 : lane L: m=L%16, half=L/16; v0=A[m][k+2*half], v1=A[m][k+2*half+1]
//   B 4x16 : lane L: n=L%16, half=L/16; v0=B[k+2*half][n], v1=B[k+2*half+1][n]
//   C/D    : lane L: n=L%16; vgpr j -> row j + 8*half
// ---------------------------------------------------------------------------
template <int FIN, int NT>
__global__ __launch_bounds__(256) void sage_gemm_wmma(
    const float* __restrict__ Agg,   // [n_dst, FIN] normalized aggregate
    const float* __restrict__ Xdst,  // [>=n_dst, FIN] destination features
    const float* __restrict__ Wl,    // [FIN, fout]
    const float* __restrict__ Wr,    // [FIN, fout]
    const float* __restrict__ bias,  // [fout]
    float* __restrict__ Out,         // [n_dst, fout]
    int n_dst, int fout, int relu)
{
    __shared__ float lWl[NT][FIN * 16];
    __shared__ float lWr[NT][FIN * 16];

    const int tilesM = n_dst >> 4;
    const int wib  = threadIdx.x >> 5;
    const int lane = threadIdx.x & 31;
    const int wid  = blockIdx.x * 8 + wib;
    const int tm   = wid % tilesM;
    const int tnBase = ((blockIdx.x * 8) / tilesM) * NT;   // block-uniform

    // ---- stage B tiles (Wl, Wr columns [tnBase*16, (tnBase+NT)*16)) ----
    for (int idx = threadIdx.x; idx < NT * FIN * 16; idx += 256) {
        const int t   = idx / (FIN * 16);
        const int rem = idx - t * (FIN * 16);
        const int kk  = rem >> 4;
        int gc = ((tnBase + t) << 4) + (rem & 15);
        if (gc >= fout) gc = fout - 1;            // clamp, mask at store
        lWl[t][rem] = Wl[(size_t)kk * fout + gc];
        lWr[t][rem] = Wr[(size_t)kk * fout + gc];
    }
    __syncthreads();

    const int half = lane >> 4;                   // K lo pair / hi pair
    const int l16  = lane & 15;
    const int m = (tm << 4) + l16;                // A row for this lane

    const float* Arow = Agg  + (size_t)m * FIN;
    const float* Xrow = Xdst + (size_t)m * FIN;

    v8f c[NT];
    #pragma unroll
    for (int t = 0; t < NT; ++t) c[t] = (v8f){};

    #pragma unroll 8
    for (int k = 0; k < FIN; k += 4) {
        const int ka = k + 2 * half;
        v2f a; a.x = Arow[ka]; a.y = Arow[ka + 1];
        #pragma unroll
        for (int t = 0; t < NT; ++t) {
            v2f b; b.x = lWl[t][ka * 16 + l16]; b.y = lWl[t][(ka + 1) * 16 + l16];
            c[t] = __builtin_amdgcn_wmma_f32_16x16x4_f32(
                       false, a, false, b, (short)0, c[t], false, false);
        }
    }
    #pragma unroll 8
    for (int k = 0; k < FIN; k += 4) {
        const int ka = k + 2 * half;
        v2f a; a.x = Xrow[ka]; a.y = Xrow[ka + 1];
        #pragma unroll
        for (int t = 0; t < NT; ++t) {
            v2f b; b.x = lWr[t][ka * 16 + l16]; b.y = lWr[t][(ka + 1) * 16 + l16];
            c[t] = __builtin_amdgcn_wmma_f32_16x16x4_f32(
                       false, a, false, b, (short)0, c[t], false, false);
        }
    }

    #pragma unroll
    for (int t = 0; t < NT; ++t) {
        const int n = ((tnBase + t) << 4) + l16;  // output column for this lane
        if (n < fout) {
            const float bv = bias[n];
            #pragma unroll
            for (int j = 0; j < 8; ++j) {
                const int row = (tm << 4) + j + (half << 3);
                float v = c[t][j] + bv;
                if (relu) v = fmaxf(v, 0.0f);
                Out[(size_t)row * fout + n] = v;
            }
        }
    }
}

// ---------------------------------------------------------------------------
// Row-wise log_softmax: one wave32 per row
// ---------------------------------------------------------------------------
__global__ __launch_bounds__(256) void log_softmax_rows(
    const float* __restrict__ in, float* __restrict__ out, int rows, int cols)
{
    const int lane = threadIdx.x & 31;
    const int r = blockIdx.x * (blockDim.x >> 5) + (threadIdx.x >> 5);
    if (r >= rows) return;
    const float* p = in + (size_t)r * cols;

    float mx = -INFINITY;
    for (int cidx = lane; cidx < cols; cidx += 32) mx = fmaxf(mx, p[cidx]);
    #pragma unroll
    for (int o = 16; o > 0; o >>= 1) mx = fmaxf(mx, __shfl_xor(mx, o, 32));

    float s = 0.0f;
    for (int cidx = lane; cidx < cols; cidx += 32) s += __expf(p[cidx] - mx);
    #pragma unroll
    for (int o = 16; o > 0; o >>= 1) s += __shfl_xor(s, o, 32);

    const float lse = __logf(s) + mx;
    for (int cidx = lane; cidx < cols; cidx += 32)
        out[(size_t)r * cols + cidx] = p[cidx] - lse;
}

// ---------------------------------------------------------------------------
// Host driver
// ---------------------------------------------------------------------------
extern "C" void kernel_launch(void* const* d_in, const int* in_sizes, int n_in,
                              void* d_out, int out_size, void* d_ws, size_t ws_size,
                              hipStream_t stream)
{
    (void)n_in; (void)out_size; (void)ws_size;

    const float* x    = (const float*)d_in[0];
    const int*   src0 = (const int*)d_in[1];
    const int*   dst0 = (const int*)d_in[2];
    const int*   src1 = (const int*)d_in[3];
    const int*   dst1 = (const int*)d_in[4];
    const int*   src2 = (const int*)d_in[5];
    const int*   dst2 = (const int*)d_in[6];
    const float* Wl0  = (const float*)d_in[7];
    const float* bl0  = (const float*)d_in[8];
    const float* Wr0  = (const float*)d_in[9];
    const float* Wl1  = (const float*)d_in[10];
    const float* bl1  = (const float*)d_in[11];
    const float* Wr1  = (const float*)d_in[12];
    const float* Wl2  = (const float*)d_in[13];
    const float* bl2  = (const float*)d_in[14];
    const float* Wr2  = (const float*)d_in[15];

    const int E0 = in_sizes[1], E1 = in_sizes[3], E2 = in_sizes[5];
    const int N1 = 65536, N2 = 8192, N3 = 1024;
    const int F0 = 100, F1 = 256, F2 = 256, F3 = 47;

    // Workspace layout (floats). h0 persists through stage 1; the scratch
    // region after h0 is reused: stage0 {agg0,cnt0} -> stage1/2 {h1,agg1,...}.
    float* ws   = (float*)d_ws;
    float* h0   = ws;                                   // 65536*256
    float* sc   = h0 + (size_t)N1 * F1;
    // stage 0
    float* agg0 = sc;                                   // 65536*100
    float* cnt0 = agg0 + (size_t)N1 * F0;               // 65536
    // stage 1/2 (reuse sc)
    float* h1     = sc;                                 // 8192*256
    float* agg1   = h1 + (size_t)N2 * F2;               // 8192*256
    float* cnt1   = agg1 + (size_t)N2 * F1;             // 8192
    float* agg2   = cnt1 + N2;                          // 1024*256
    float* cnt2   = agg2 + (size_t)N3 * F2;             // 1024
    float* logits = cnt2 + N3;                          // 1024*47

    const int THREADS = 256;
    const int WPB = THREADS / 32;

    // ---------------- Layer 0: x(524288x100) -> h0(65536x256), relu ----------
    {
        long nz = (long)N1 * F0 + N1;
        fill_zero_f32<<<2048, THREADS, 0, stream>>>(agg0, nz);
        sage_scatter<<<(E0 + WPB - 1) / WPB, THREADS, 0, stream>>>(
            x, src0, dst0, agg0, cnt0, E0, F0);
        sage_normalize<<<4096, THREADS, 0, stream>>>(agg0, cnt0, (long)N1 * F0, F0);
        // tiles: M=4096, Ngroups=16/2=8 -> 32768 waves -> 4096 blocks
        int waves = (N1 / 16) * ((F1 / 16) / 2);
        sage_gemm_wmma<F0, 2><<<waves / WPB, THREADS, 0, stream>>>(
            agg0, x, Wl0, Wr0, bl0, h0, N1, F1, /*relu=*/1);
    }

    // ---------------- Layer 1: h0(65536x256) -> h1(8192x256), relu -----------
    {
        long nz = (long)N2 * F1 + N2;
        fill_zero_f32<<<2048, THREADS, 0, stream>>>(agg1, nz);
        sage_scatter<<<(E1 + WPB - 1) / WPB, THREADS, 0, stream>>>(
            h0, src1, dst1, agg1, cnt1, E1, F1);
        sage_normalize<<<2048, THREADS, 0, stream>>>(agg1, cnt1, (long)N2 * F1, F1);
        // tiles: M=512, Ngroups=8 -> 4096 waves -> 512 blocks
        int waves = (N2 / 16) * ((F2 / 16) / 2);
        sage_gemm_wmma<F1, 2><<<waves / WPB, THREADS, 0, stream>>>(
            agg1, h0, Wl1, Wr1, bl1, h1, N2, F2, /*relu=*/1);
    }

    // ---------------- Layer 2: h1(8192x256) -> logits(1024x47) ---------------
    {
        long nz = (long)N3 * F2 + N3;
        fill_zero_f32<<<512, THREADS, 0, stream>>>(agg2, nz);
        sage_scatter<<<(E2 + WPB - 1) / WPB, THREADS, 0, stream>>>(
            h1, src2, dst2, agg2, cnt2, E2, F2);
        sage_normalize<<<512, THREADS, 0, stream>>>(agg2, cnt2, (long)N3 * F2, F2);
        // tiles: M=64, Ngroups=3 (fout=47, NT=1) -> 192 waves -> 24 blocks
        int waves = (N3 / 16) * ((F3 + 15) / 16);
        sage_gemm_wmma<F2, 1><<<waves / WPB, THREADS, 0, stream>>>(
            agg2, h1, Wl2, Wr2, bl2, logits, N3, F3, /*relu=*/0);
    }

    // ---------------- log_softmax over rows of logits -> d_out ---------------
    log_softmax_rows<<<(N3 + WPB - 1) / WPB, THREADS, 0, stream>>>(
        logits, (float*)d_out, N3, F3);
}